// QuantumVQCHead_28269474743038
// MI455X (gfx1250) — compile-verified
//
#include <hip/hip_runtime.h>
#include <cstddef>

// ---------------------------------------------------------------------------
// QuantumVQCHead for MI455X (gfx1250, wave32)
//   Kernel 0: precompute fused per-qubit U = RZ*RY*RX (batch-uniform) -> ws
//   Kernel 1: a = clip(tanh(h @ W_pre.T + b) * pi/2)  via V_WMMA_F32_16X16X4_F32
//             K split 4-ways per tile + LDS reduction for memory-level par.
//   Kernel 2: 8-qubit statevector circuit, 1 batch element per wave,
//             8 complex amps per lane, shfl_xor for cross-lane qubits,
//             CNOT chain folded into compile-time linear masks.
// ---------------------------------------------------------------------------

typedef __attribute__((ext_vector_type(2))) float v2f;
typedef __attribute__((ext_vector_type(8))) float v8f;

#define PI_F 3.14159265358979323846f
#define KSPLIT 4

// ----------------------- compile-time CNOT algebra -------------------------
__host__ __device__ constexpr int cnot_perm(int i, int c, int t) {
  return i ^ (((i >> (7 - c)) & 1) << (7 - t));
}
// state_final[i] = state0[P(i)],  P = p(0,1) o p(1,2) o ... o p(6,7) o p(7,0)
__host__ __device__ constexpr int Pfun(int i) {
  int x = i;
  x = cnot_perm(x, 7, 0);
  x = cnot_perm(x, 6, 7);
  x = cnot_perm(x, 5, 6);
  x = cnot_perm(x, 4, 5);
  x = cnot_perm(x, 3, 4);
  x = cnot_perm(x, 2, 3);
  x = cnot_perm(x, 1, 2);
  x = cnot_perm(x, 0, 1);
  return x;
}
__host__ __device__ constexpr int Pinv(int j) {
  for (int i = 0; i < 256; ++i)
    if (Pfun(i) == j) return i;
  return 0;
}
__host__ __device__ constexpr int flip_mask(int w) { return Pfun(1 << (7 - w)); }
__host__ __device__ constexpr int sign_mask(int w) {
  int m = 0;
  for (int k = 0; k < 8; ++k)
    if ((Pinv(1 << k) >> (7 - w)) & 1) m |= 1 << k;
  return m;
}

// ----------------------- fast branch-free tanh -----------------------------
__device__ __forceinline__ float fast_tanh(float x) {
#if __has_builtin(__builtin_amdgcn_tanhf)
  return __builtin_amdgcn_tanhf(x);
#elif __has_builtin(__builtin_amdgcn_tanh_f32)
  return __builtin_amdgcn_tanh_f32(x);
#else
  const float e = __expf(2.0f * x);
  return 1.0f - 2.0f / (e + 1.0f);
#endif
}

// ---------------------------------------------------------------------------
// Kernel 0: fused per-qubit gate U = RZ*RY*RX  -> Uws[8][8]
//   layout per qubit: u00r,u00i,u01r,u01i,u10r,u10i,u11r,u11i
// ---------------------------------------------------------------------------
__global__ void vqc_prep_gates(const float* __restrict__ Wts,
                               float* __restrict__ Uws) {
  const int w = threadIdx.x;
  if (w < 8) {
    const float tx = Wts[w * 3 + 0] * 0.5f;
    const float ty = Wts[w * 3 + 1] * 0.5f;
    const float tz = Wts[w * 3 + 2] * 0.5f;
    const float cx = __cosf(tx), sx = __sinf(tx);
    const float cy = __cosf(ty), sy = __sinf(ty);
    const float cz = __cosf(tz), sz = __sinf(tz);
    // RY*RX = [[A+iB, -C-iD],[C-iD, A-iB]], A=cy*cx B=sy*sx C=sy*cx D=cy*sx
    const float Am = cy * cx, Bm = sy * sx, Cm = sy * cx, Dm = cy * sx;
    float* o = Uws + w * 8;
    o[0] = cz * Am + sz * Bm;   // u00r   (row0 *= cz - i sz)
    o[1] = cz * Bm - sz * Am;   // u00i
    o[2] = -cz * Cm - sz * Dm;  // u01r
    o[3] = -cz * Dm + sz * Cm;  // u01i
    o[4] = cz * Cm + sz * Dm;   // u10r   (row1 *= cz + i sz)
    o[5] = sz * Cm - cz * Dm;   // u10i
    o[6] = cz * Am + sz * Bm;   // u11r
    o[7] = sz * Am - cz * Bm;   // u11i
  }
}

// ---------------------------------------------------------------------------
// Kernel 1: WMMA GEMM (Bx512) x (512x8) + bias + tanh*pi/2 + clip
// One wave computes a 16x16 tile over a 128-wide K section; 4 waves per tile
// reduce partials through LDS.  Block = 256 thr = 8 waves = 2 tiles.
// ---------------------------------------------------------------------------
__global__ void __launch_bounds__(256) vqc_gemm_act(
    const float* __restrict__ h,     // [B,512]
    const float* __restrict__ Wp,    // [8,512]
    const float* __restrict__ bp,    // [8]
    float* __restrict__ Aout)        // [B,8]
{
  __shared__ float red[2][KSPLIT - 1][256];

  const int lane = threadIdx.x & 31;
  const int wv = threadIdx.x >> 5;       // 0..7
  const int tileInBlk = wv >> 2;         // 0..1
  const int ksec = wv & (KSPLIT - 1);    // 0..3
  const int tile = blockIdx.x * 2 + tileInBlk;
  const int rowBase = tile * 16;

  const int half = lane >> 4;            // 0: K={0,1}  1: K={2,3} per step
  const int l = lane & 15;               // A: M index, B: N index
  const int koff = half * 2;
  const int kbase = ksec * (512 / KSPLIT);

  const float* hrow = h + (size_t)(rowBase + l) * 512 + kbase;
  const bool nvalid = (l < 8);
  const float* wrow = Wp + (size_t)(nvalid ? l : 0) * 512 + kbase;
  const float wmask = nvalid ? 1.0f : 0.0f;

  v8f acc = {};
#pragma unroll 8
  for (int k = 0; k < 512 / KSPLIT; k += 4) {
    v2f a, b;
    a.x = hrow[k + koff];
    a.y = hrow[k + koff + 1];
    b.x = wrow[k + koff] * wmask;
    b.y = wrow[k + koff + 1] * wmask;
    acc = __builtin_amdgcn_wmma_f32_16x16x4_f32(
        /*neg_a=*/false, a, /*neg_b=*/false, b,
        /*c_mod=*/(short)0, acc, /*reuse_a=*/false, /*reuse_b=*/false);
  }

  // partial reduction across the 4 K-sections of this tile
  if (ksec != 0) {
    float* dst = &red[tileInBlk][ksec - 1][lane * 8];
#pragma unroll
    for (int r = 0; r < 8; ++r) dst[r] = acc[r];
  }
  __syncthreads();

  if (ksec == 0) {
#pragma unroll
    for (int p = 0; p < KSPLIT - 1; ++p) {
      const float* srcp = &red[tileInBlk][p][lane * 8];
#pragma unroll
      for (int r = 0; r < 8; ++r) acc[r] += srcp[r];
    }
    // C/D layout: VGPR r -> M = r + 8*half, N = l
    if (nvalid) {
      const float bias = bp[l];
#pragma unroll
      for (int r = 0; r < 8; ++r) {
        float t = acc[r] + bias;
        float v = fast_tanh(t) * (PI_F * 0.5f);
        v = fminf(fmaxf(v, -PI_F), PI_F);
        Aout[(size_t)(rowBase + r + 8 * half) * 8 + l] = v;
      }
    }
  }
}

// ---------------------------------------------------------------------------
// Kernel 2: circuit simulation.  One batch element per wave (wave32).
// Lane holds amplitudes j = lane*8 + s, s = 0..7.
// Index bit b: b in {0,1,2} -> slot bit, b in {3..7} -> lane bit (b-3).
// ---------------------------------------------------------------------------
__global__ void __launch_bounds__(256) vqc_circuit(
    const float* __restrict__ Ain,   // [B,8]  angles
    const float* __restrict__ Uws,   // [8][8] fused gate matrices
    float* __restrict__ out)         // [B,24]
{
  const int lane = threadIdx.x & 31;
  const int elem = (int)((blockIdx.x * blockDim.x + threadIdx.x) >> 5);
  const float* a = Ain + (size_t)elem * 8;

  // ---- product-state init from the 8 batch-dependent RY gates on |0..0> ---
  float cw[8], sw[8];
#pragma unroll
  for (int w = 0; w < 8; ++w) {
    float t = a[w] * 0.5f;
    cw[w] = __cosf(t);
    sw[w] = __sinf(t);
  }
  // qubits 0..4 live on lane bits 4..0; qubits 5..7 on slot bits 2..0
  float base = 1.0f;
#pragma unroll
  for (int w = 0; w < 5; ++w)
    base *= ((lane >> (4 - w)) & 1) ? sw[w] : cw[w];

  float sp[8];
  {
    const float p00 = cw[5] * cw[6], p01 = cw[5] * sw[6];
    const float p10 = sw[5] * cw[6], p11 = sw[5] * sw[6];
    sp[0] = p00 * cw[7]; sp[1] = p00 * sw[7];
    sp[2] = p01 * cw[7]; sp[3] = p01 * sw[7];
    sp[4] = p10 * cw[7]; sp[5] = p10 * sw[7];
    sp[6] = p11 * cw[7]; sp[7] = p11 * sw[7];
  }
  float sr[8], si[8];
#pragma unroll
  for (int s = 0; s < 8; ++s) {
    sr[s] = base * sp[s];
    si[s] = 0.0f;
  }

  // ---- one layer: per qubit fused U = RZ * RY * RX (precomputed) ----------
#pragma unroll
  for (int w = 0; w < 8; ++w) {
    const float* U = Uws + w * 8;
    const float u00r = U[0], u00i = U[1];
    const float u01r = U[2], u01i = U[3];
    const float u10r = U[4], u10i = U[5];
    const float u11r = U[6], u11i = U[7];

    const int b = 7 - w;  // index bit this qubit acts on
    if (b >= 3) {
      // cross-lane pairing via shfl_xor
      const int lm = 1 << (b - 3);
      const int bit = (lane >> (b - 3)) & 1;
      const float c0r = bit ? u11r : u00r;  // coeff on my amp
      const float c0i = bit ? u11i : u00i;
      const float c1r = bit ? u10r : u01r;  // coeff on partner amp
      const float c1i = bit ? u10i : u01i;
#pragma unroll
      for (int s = 0; s < 8; ++s) {
        const float pr = __shfl_xor(sr[s], lm, 32);
        const float pi = __shfl_xor(si[s], lm, 32);
        const float mr = sr[s], mi = si[s];
        sr[s] = c0r * mr - c0i * mi + c1r * pr - c1i * pi;
        si[s] = c0r * mi + c0i * mr + c1r * pi + c1i * pr;
      }
    } else {
      // in-lane pairing: slots (s0, s0|sm)
      const int sm = 1 << b;
#pragma unroll
      for (int s0 = 0; s0 < 8; ++s0) {
        if (s0 & sm) continue;
        const int s1 = s0 | sm;
        const float x0r = sr[s0], x0i = si[s0];
        const float x1r = sr[s1], x1i = si[s1];
        sr[s0] = u00r * x0r - u00i * x0i + u01r * x1r - u01i * x1i;
        si[s0] = u00r * x0i + u00i * x0r + u01r * x1i + u01i * x1r;
        sr[s1] = u10r * x0r - u10i * x0i + u11r * x1r - u11i * x1i;
        si[s1] = u10r * x0i + u10i * x0r + u11r * x1i + u11i * x1r;
      }
    }
  }

  // ---- measurement with CNOT chain folded into compile-time masks ---------
  constexpr int FW[8] = {flip_mask(0), flip_mask(1), flip_mask(2), flip_mask(3),
                         flip_mask(4), flip_mask(5), flip_mask(6), flip_mask(7)};
  constexpr int MW[8] = {sign_mask(0), sign_mask(1), sign_mask(2), sign_mask(3),
                         sign_mask(4), sign_mask(5), sign_mask(6), sign_mask(7)};

  float outv[24];
#pragma unroll
  for (int w = 0; w < 8; ++w) {
    const int F = FW[w];
    const int slotX = F & 7;
    const int laneX = F >> 3;
    const int mk = MW[w];
    float a0 = 0.0f, a1 = 0.0f, a2 = 0.0f;
#pragma unroll
    for (int s = 0; s < 8; ++s) {
      float pr = sr[s ^ slotX];
      float pi = si[s ^ slotX];
      if (laneX) {  // compile-time constant branch
        pr = __shfl_xor(pr, laneX, 32);
        pi = __shfl_xor(pi, laneX, 32);
      }
      const float mr = sr[s], mi = si[s];
      const float reo = mr * pr + mi * pi;   // Re(conj(m)*p)
      const float imo = mr * pi - mi * pr;   // Im(conj(m)*p)
      const float prob = mr * mr + mi * mi;
      const int j = (lane << 3) | s;
      const float sgn = (__popc(j & mk) & 1) ? -1.0f : 1.0f;
      a0 += reo;
      a1 += sgn * imo;
      a2 += sgn * prob;
    }
    outv[w * 3 + 0] = a0;
    outv[w * 3 + 1] = a1;
    outv[w * 3 + 2] = a2;
  }

  // ---- wave butterfly reduction of the 24 sums ----------------------------
#pragma unroll
  for (int t = 0; t < 24; ++t) {
    float v = outv[t];
    v += __shfl_xor(v, 1, 32);
    v += __shfl_xor(v, 2, 32);
    v += __shfl_xor(v, 4, 32);
    v += __shfl_xor(v, 8, 32);
    v += __shfl_xor(v, 16, 32);
    outv[t] = v;
  }

  float nrm2 = 0.0f;
#pragma unroll
  for (int t = 0; t < 24; ++t) nrm2 += outv[t] * outv[t];
  const float inv = 1.0f / fmaxf(__builtin_sqrtf(nrm2), 1e-12f);

  if (lane == 0) {
    float* o = out + (size_t)elem * 24;
#pragma unroll
    for (int t = 0; t < 24; ++t) o[t] = outv[t] * inv;
  }
}

// ---------------------------------------------------------------------------
extern "C" void kernel_launch(void* const* d_in, const int* in_sizes, int n_in,
                              void* d_out, int out_size, void* d_ws,
                              size_t ws_size, hipStream_t stream) {
  const float* h = (const float*)d_in[0];    // [B,512]
  const float* Wp = (const float*)d_in[1];   // [8,512]
  const float* bp = (const float*)d_in[2];   // [8]
  const float* wt = (const float*)d_in[3];   // [1,8,3]
  float* out = (float*)d_out;                // [B,24]

  float* Uws = (float*)d_ws;                 // [8*8] fused gate matrices
  float* Aws = (float*)d_ws + 64;            // [B,8] angles scratch

  const int B = in_sizes[0] / 512;           // 16384

  // Kernel 0: fused gate matrices (uniform across batch)
  vqc_prep_gates<<<1, 32, 0, stream>>>(wt, Uws);

  // Kernel 1: 4 waves per 16-row tile (K split), 2 tiles per 256-thr block
  const int tiles = B / 16;
  vqc_gemm_act<<<tiles / 2, 256, 0, stream>>>(h, Wp, bp, Aws);

  // Kernel 2: 1 wave per batch element, 8 waves per block
  vqc_circuit<<<B / 8, 256, 0, stream>>>(Aws, Uws, out);
}